// SPADE_37495064494581
// MI455X (gfx1250) — compile-verified
//
#include <hip/hip_runtime.h>
#include <hip/hip_bf16.h>

typedef __attribute__((ext_vector_type(2))) float v2f;
typedef __attribute__((ext_vector_type(8))) float v8f;

#define BATCH 8
#define NLIB 200
#define DIM 512
#define TOPK 5
#define NTILES 13           // ceil(200/16)
#define NPAD 208            // 13*16
#define IMG 224

// ---------------- workspace layout (float offsets) ----------------
#define WS_ZNORM   0        // 8
#define WS_LIBNORM 8        // 200
#define WS_DIST    256      // 8*208
#define WS_IDX     1984     // 40 ints
#define WS_SMAP0   2048     // 8*3136
#define WS_SMAP1   27136    // 8*784
#define WS_SMAP2   33408    // 8*196

// ---------------- kernel 0: row norms (float4 streamed) ----------------
__global__ void spade_norms(const float* __restrict__ z,
                            const float* __restrict__ zlib,
                            float* __restrict__ ws) {
    int i = blockIdx.x * blockDim.x + threadIdx.x;
    const float* r;
    float* dst;
    if (i < BATCH)            { r = z    + (size_t)i * DIM;          dst = ws + WS_ZNORM + i; }
    else if (i < BATCH + NLIB){ r = zlib + (size_t)(i - BATCH) * DIM; dst = ws + WS_LIBNORM + (i - BATCH); }
    else return;
    const float4* r4 = (const float4*)r;
    float s = 0.f;
    #pragma unroll 4
    for (int d = 0; d < DIM / 4; ++d) {
        float4 v = r4[d];
        s = fmaf(v.x, v.x, s); s = fmaf(v.y, v.y, s);
        s = fmaf(v.z, v.z, s); s = fmaf(v.w, v.w, s);
    }
    *dst = s;
}

// ---------------- kernel 1: distance matrix via f32 WMMA ----------------
// One wave per 16-column tile. Rows m>=8 / cols n>=200 use clamped (valid)
// addresses and produce garbage ONLY in output elements we never read, so
// loads are unconditional b64 -> no exec-mask branching in the hot loop.
__global__ __launch_bounds__(NTILES * 32)
void spade_dist_wmma(const float* __restrict__ z,
                     const float* __restrict__ zlib,
                     float* __restrict__ ws) {
    const int wave = threadIdx.x >> 5;         // 0..12 -> column tile
    const int lane = threadIdx.x & 31;
    const int m    = lane & 15;                // A row this lane carries
    const int kh   = lane >> 4;                // K half-pair selector (0/1)
    const int n    = wave * 16 + (lane & 15);  // B column this lane carries

    const float* ap = z    + (size_t)(m & 7) * DIM            + kh * 2;
    const float* bp = zlib + (size_t)(n < NLIB ? n : 0) * DIM + kh * 2;

    v8f c0 = {}, c1 = {};
    #pragma unroll 4
    for (int k0 = 0; k0 < DIM; k0 += 8) {
        v2f a0 = *(const v2f*)(ap + k0);
        v2f b0 = *(const v2f*)(bp + k0);
        v2f a1 = *(const v2f*)(ap + k0 + 4);
        v2f b1 = *(const v2f*)(bp + k0 + 4);
        c0 = __builtin_amdgcn_wmma_f32_16x16x4_f32(
                 false, a0, false, b0, (short)0, c0, false, false);
        c1 = __builtin_amdgcn_wmma_f32_16x16x4_f32(
                 false, a1, false, b1, (short)0, c1, false, false);
    }
    v8f c = c0 + c1;

    // C/D layout: lanes 0-15 hold N=lane, VGPR r -> M=r (rows 0..7 = batch)
    if (lane < 16) {
        int nn = wave * 16 + lane;
        if (nn < NLIB) {
            float ln = ws[WS_LIBNORM + nn];
            #pragma unroll
            for (int r = 0; r < BATCH; ++r) {
                float d2 = ws[WS_ZNORM + r] + ln - 2.0f * c[r];
                ws[WS_DIST + r * NPAD + nn] = sqrtf(fmaxf(d2, 0.0f));
            }
        }
    }
}

// ---------------- kernel 2: top-k smallest + z_score ----------------
__global__ void spade_topk(const float* __restrict__ ws_dist,
                           float* __restrict__ zscore_out,
                           int* __restrict__ idx_out) {
    int b = threadIdx.x;
    if (b >= BATCH) return;
    const float* d = ws_dist + b * NPAD;
    int chosen[TOPK];
    float sum = 0.f;
    #pragma unroll
    for (int j = 0; j < TOPK; ++j) {
        float best = 3.4e38f;
        int bi = 0;
        for (int nidx = 0; nidx < NLIB; ++nidx) {
            bool skip = false;
            #pragma unroll
            for (int t = 0; t < TOPK; ++t)
                if (t < j && chosen[t] == nidx) skip = true;
            float v = d[nidx];
            if (!skip && v < best) { best = v; bi = nidx; }
        }
        chosen[j] = bi;
        sum += best;
    }
    zscore_out[b] = sum * (1.0f / TOPK);
    #pragma unroll
    for (int j = 0; j < TOPK; ++j) idx_out[b * TOPK + j] = chosen[j];
}

// ---------------- kernel 3: per-level anomaly map ----------------
// smap[b][p] = min_j sum_c (lib[idx[b][j]][c][p] - fmap[b][c][p])^2
// All 5 neighbors fused into the channel loop: 6 independent float4 loads
// per iteration per lane (max memory-level parallelism), fmap streamed once.
// Software prefetch (no LOADcnt slot used) covers the HW*4-byte channel
// stride of the 5 gathered lib streams.
__global__ void spade_smap(const float* __restrict__ fmap,
                           const float* __restrict__ lib,
                           const int* __restrict__ idx,
                           float* __restrict__ smap,
                           int C, int HW) {
    int b   = blockIdx.y;
    int HW4 = HW >> 2;
    int q   = blockIdx.x * blockDim.x + threadIdx.x;   // pixel-quad index
    if (q >= HW4) return;

    const float4* f = (const float4*)(fmap + (size_t)b * C * HW);
    const float4* l[TOPK];
    #pragma unroll
    for (int j = 0; j < TOPK; ++j)
        l[j] = (const float4*)(lib + (size_t)idx[b * TOPK + j] * C * HW);

    float4 acc[TOPK];
    #pragma unroll
    for (int j = 0; j < TOPK; ++j)
        acc[j].x = acc[j].y = acc[j].z = acc[j].w = 0.f;

    for (int c = 0; c < C; ++c) {
        size_t off = (size_t)c * HW4 + q;
        if (c + 8 < C) {
            size_t poff = (size_t)(c + 8) * HW4 + q;
            #pragma unroll
            for (int j = 0; j < TOPK; ++j)
                __builtin_prefetch((const void*)(l[j] + poff), 0, 0);
        }
        float4 fv = f[off];
        #pragma unroll
        for (int j = 0; j < TOPK; ++j) {
            float4 lv = l[j][off];
            float dx = lv.x - fv.x, dy = lv.y - fv.y;
            float dz = lv.z - fv.z, dw = lv.w - fv.w;
            acc[j].x = fmaf(dx, dx, acc[j].x);
            acc[j].y = fmaf(dy, dy, acc[j].y);
            acc[j].z = fmaf(dz, dz, acc[j].z);
            acc[j].w = fmaf(dw, dw, acc[j].w);
        }
    }

    float4 best = acc[0];
    #pragma unroll
    for (int j = 1; j < TOPK; ++j) {
        best.x = fminf(best.x, acc[j].x); best.y = fminf(best.y, acc[j].y);
        best.z = fminf(best.z, acc[j].z); best.w = fminf(best.w, acc[j].w);
    }
    ((float4*)(smap + (size_t)b * HW))[q] = best;
}

// ---------------- kernel 4: bilinear upsample (half-pixel) + sum ----------------
__device__ __forceinline__ float bil_sample(const float* __restrict__ s, int H,
                                            int y, int x) {
    float scale = (float)H / (float)IMG;
    float sy = ((float)y + 0.5f) * scale - 0.5f;
    float sx = ((float)x + 0.5f) * scale - 0.5f;
    int y0 = (int)floorf(sy), x0 = (int)floorf(sx);
    float fy = sy - (float)y0, fx = sx - (float)x0;
    int y1 = y0 + 1, x1 = x0 + 1;
    y0 = min(max(y0, 0), H - 1); y1 = min(max(y1, 0), H - 1);
    x0 = min(max(x0, 0), H - 1); x1 = min(max(x1, 0), H - 1);
    float v00 = s[y0 * H + x0], v01 = s[y0 * H + x1];
    float v10 = s[y1 * H + x0], v11 = s[y1 * H + x1];
    return v00 * (1.f - fy) * (1.f - fx) + v01 * (1.f - fy) * fx
         + v10 * fy * (1.f - fx) + v11 * fy * fx;
}

__global__ void spade_upsample(const float* __restrict__ s0,
                               const float* __restrict__ s1,
                               const float* __restrict__ s2,
                               float* __restrict__ out_map) {
    int b = blockIdx.y;
    int t = blockIdx.x * blockDim.x + threadIdx.x;
    if (t >= IMG * IMG) return;
    int y = t / IMG, x = t % IMG;
    float acc = bil_sample(s0 + (size_t)b * 56 * 56, 56, y, x)
              + bil_sample(s1 + (size_t)b * 28 * 28, 28, y, x)
              + bil_sample(s2 + (size_t)b * 14 * 14, 14, y, x);
    out_map[(size_t)b * IMG * IMG + t] = acc;
}

extern "C" void kernel_launch(void* const* d_in, const int* in_sizes, int n_in,
                              void* d_out, int out_size, void* d_ws, size_t ws_size,
                              hipStream_t stream) {
    (void)in_sizes; (void)n_in; (void)out_size; (void)ws_size;
    const float* z     = (const float*)d_in[0];
    const float* z_lib = (const float*)d_in[1];
    const float* fmap0 = (const float*)d_in[2];
    const float* fmap1 = (const float*)d_in[3];
    const float* fmap2 = (const float*)d_in[4];
    const float* lib0  = (const float*)d_in[5];
    const float* lib1  = (const float*)d_in[6];
    const float* lib2  = (const float*)d_in[7];
    // d_in[8] is k (==5), fixed at compile time as TOPK.

    float* ws   = (float*)d_ws;
    float* out  = (float*)d_out;            // [0,8): z_score ; [8,...): map
    int*   idxp = (int*)(ws + WS_IDX);

    // 1) norms
    spade_norms<<<1, 256, 0, stream>>>(z, z_lib, ws);
    // 2) distances via f32 WMMA (13 waves = 13 column tiles)
    spade_dist_wmma<<<1, NTILES * 32, 0, stream>>>(z, z_lib, ws);
    // 3) top-k + z_score (writes d_out[0..7]) + indices
    spade_topk<<<1, 32, 0, stream>>>(ws + WS_DIST, out, idxp);
    // 4) per-level min-distance maps (5 neighbors fused, float4 per lane)
    {
        dim3 g0(((56 * 56 / 4) + 255) / 256, BATCH);
        spade_smap<<<g0, 256, 0, stream>>>(fmap0, lib0, idxp, ws + WS_SMAP0, 64, 56 * 56);
        dim3 g1(((28 * 28 / 4) + 255) / 256, BATCH);
        spade_smap<<<g1, 256, 0, stream>>>(fmap1, lib1, idxp, ws + WS_SMAP1, 128, 28 * 28);
        dim3 g2(((14 * 14 / 4) + 255) / 256, BATCH);
        spade_smap<<<g2, 256, 0, stream>>>(fmap2, lib2, idxp, ws + WS_SMAP2, 256, 14 * 14);
    }
    // 5) upsample+sum into d_out[8 ...]
    {
        dim3 gu((IMG * IMG + 255) / 256, BATCH);
        spade_upsample<<<gu, 256, 0, stream>>>(ws + WS_SMAP0, ws + WS_SMAP1,
                                               ws + WS_SMAP2, out + BATCH);
    }
}